// RobertaSelfAttention_matchKV_283467842267
// MI455X (gfx1250) — compile-verified
//
#include <hip/hip_runtime.h>

#define BSZ  4
#define LSEQ 2048
#define NH   12
#define HDIM 64
#define RR   16
#define HIDD 768
#define BPAD 40   // LDS row stride for B panel (2*40 % 64 == 16 -> conflict-free halves)

typedef __attribute__((ext_vector_type(2))) float v2f;
typedef __attribute__((ext_vector_type(4))) float v4f;
typedef __attribute__((ext_vector_type(8))) float v8f;

__device__ __forceinline__ v8f wmma_f32(v2f a, v2f b, v8f c) {
  // V_WMMA_F32_16X16X4_F32 : D = A(16x4,f32) * B(4x16,f32) + C(16x16,f32)
  return __builtin_amdgcn_wmma_f32_16x16x4_f32(
      /*neg_a=*/false, a, /*neg_b=*/false, b,
      /*c_mod=*/(short)0, c, /*reuse_a=*/false, /*reuse_b=*/false);
}

// Y = relu(X @ W + bias); X:(8192,768) W:(768,768) Y:(8192,768)
// Block: 8 waves, shared 32-col N-slice of B staged in LDS (768x32, stride 40).
// Wave: 64x32 output (4 M-subtiles x 2 N-subtiles), 8 WMMA accumulators.
__global__ __launch_bounds__(256) void gemm_relu_wmma(
    const float* __restrict__ X, const float* __restrict__ W,
    const float* __restrict__ bias, float* __restrict__ Y) {
  __shared__ float Bs[HIDD * BPAD];                 // 120 KB

  const int tid  = threadIdx.x;
  const int lane = tid & 31;
  const int wv   = tid >> 5;                        // 0..7
  const int nblk = blockIdx.x % (HIDD / 32);        // 0..23
  const int mblk = blockIdx.x / (HIDD / 32);        // 0..15
  const int n0   = nblk * 32;
  const int m0   = mblk * 512 + wv * 64;
  const int half = lane >> 4, lr = lane & 15;

  // Cooperative stage of B(768 x 32) into LDS (row-major, padded stride BPAD).
  // Each thread: 24 x (global_load_b128 -> ds_store_b128).
  for (int i = 0; i < 24; ++i) {
    const int e = tid + i * 256;                    // 0..6143
    const int k = e >> 3;                           // 0..767
    const int c = (e & 7) * 4;                      // 0..28
    const v4f w4 = *(const v4f*)(W + (size_t)k * HIDD + n0 + c);
    *(v4f*)(&Bs[k * BPAD + c]) = w4;
  }
  __syncthreads();

  const float* A0 = X + (size_t)(m0 + lr) * HIDD;   // M-subtile rows: +16*ti*HIDD

  v8f acc[4][2] = {};
  #pragma unroll 2
  for (int k0 = 0; k0 < HIDD; k0 += 4) {
    const int kk = k0 + 2 * half;                   // lanes 16-31 hold K+2 pair
    v2f a[4];
    #pragma unroll
    for (int ti = 0; ti < 4; ++ti)
      a[ti] = *(const v2f*)(A0 + (size_t)ti * 16 * HIDD + kk);
    v2f b0, b1;
    b0.x = Bs[kk * BPAD + lr];
    b0.y = Bs[(kk + 1) * BPAD + lr];
    b1.x = Bs[kk * BPAD + 16 + lr];
    b1.y = Bs[(kk + 1) * BPAD + 16 + lr];
    #pragma unroll
    for (int ti = 0; ti < 4; ++ti) {
      acc[ti][0] = wmma_f32(a[ti], b0, acc[ti][0]);
      acc[ti][1] = wmma_f32(a[ti], b1, acc[ti][1]);
    }
  }

  // Epilogue: C/D layout -> lane holds rows v+8*half, col = lane%16 per subtile
  const float bj0 = bias[n0 + lr];
  const float bj1 = bias[n0 + 16 + lr];
  #pragma unroll
  for (int ti = 0; ti < 4; ++ti) {
    const int rb = m0 + ti * 16 + 8 * half;
    #pragma unroll
    for (int v = 0; v < 8; ++v) {
      float y;
      y = acc[ti][0][v] + bj0;
      Y[(size_t)(rb + v) * HIDD + n0 + lr] = y > 0.f ? y : 0.f;
      y = acc[ti][1][v] + bj1;
      Y[(size_t)(rb + v) * HIDD + n0 + 16 + lr] = y > 0.f ? y : 0.f;
    }
  }
}

// dp[row,h] = sum_d K1[row, h*64+d] * RH[h*64+d]; one wave per row (b*L+l)
__global__ __launch_bounds__(256) void dp_kernel(
    const float* __restrict__ K1, const float* __restrict__ RH,
    float* __restrict__ dp) {
  const int lane = threadIdx.x & 31;
  const int row = blockIdx.x * 8 + (threadIdx.x >> 5); // 0..8191
  const float* kr = K1 + (size_t)row * HIDD;
  for (int h = 0; h < NH; ++h) {
    v2f kv = *(const v2f*)(kr + h * HDIM + 2 * lane);
    v2f rv = *(const v2f*)(RH + h * HDIM + 2 * lane);
    float p = kv.x * rv.x + kv.y * rv.y;
    #pragma unroll
    for (int s = 16; s >= 1; s >>= 1) p += __shfl_xor(p, s, 32);
    if (lane == 0) dp[(size_t)row * NH + h] = p;
  }
}

// Compact sorted valid positions per (b,h). Position 0 is never pushed
// (matches reference: scan starts at idx 1). 48 independent sequences.
__global__ void scan_kernel(const float* __restrict__ dp,
                            int* __restrict__ pos, int* __restrict__ cnt) {
  const int t = threadIdx.x;
  if (t >= BSZ * NH) return;
  const int b = t / NH, h = t % NH;
  int* p = pos + (size_t)t * LSEQ;
  int c = 0;
  for (int l = 1; l < LSEQ; ++l)
    if (dp[((size_t)b * LSEQ + l) * NH + h] > 0.5f) p[c++] = l;
  cnt[t] = c;
}

// ctx[b,l,h,:] = sum_r wF[h,r]*V1[b,fm(r),h,:] + wB[h,r]*V1[b,bm(r),h,:]
// fm(r) = r-th most recent valid index <= l (pad 0); bm(r) = r-th nearest
// valid index >= l (pad 0); l==0 -> all indices 0 (reference prepends zeros).
__global__ __launch_bounds__(256) void gather_kernel(
    const float* __restrict__ V1, const int* __restrict__ pos,
    const int* __restrict__ cnt, const float* __restrict__ bw,
    float* __restrict__ out) {
  const int lane = threadIdx.x & 31;
  const int w = blockIdx.x * 8 + (threadIdx.x >> 5); // 0..98303
  const int h = w % NH;
  const int row = w / NH; // b*L + l
  const int b = row / LSEQ;
  const int l = row % LSEQ;
  const int bh = b * NH + h;
  const int* p = pos + (size_t)bh * LSEQ;
  const int n = cnt[bh];

  // upper_bound: first idx with p[idx] > l
  int lo = 0, hi = n;
  while (lo < hi) { int mid = (lo + hi) >> 1; if (p[mid] <= l) lo = mid + 1; else hi = mid; }
  const int kU = lo;
  // lower_bound: first idx with p[idx] >= l
  lo = 0; hi = n;
  while (lo < hi) { int mid = (lo + hi) >> 1; if (p[mid] < l) lo = mid + 1; else hi = mid; }
  const int kL = lo;

  const float* Vb = V1 + (size_t)b * LSEQ * HIDD + h * HDIM + 2 * lane;
  float ax = 0.f, ay = 0.f;
  #pragma unroll
  for (int r = 0; r < RR; ++r) {
    const int j = kU - 1 - r;
    const int idx = (j >= 0) ? p[j] : 0;
    const float wgt = bw[h * (2 * RR) + r];
    v2f v = *(const v2f*)(Vb + (size_t)idx * HIDD);
    ax += wgt * v.x; ay += wgt * v.y;
  }
  #pragma unroll
  for (int r = 0; r < RR; ++r) {
    const int j = kL + r;
    const int idx = (l > 0 && j < n) ? p[j] : 0;
    const float wgt = bw[h * (2 * RR) + RR + r];
    v2f v = *(const v2f*)(Vb + (size_t)idx * HIDD);
    ax += wgt * v.x; ay += wgt * v.y;
  }
  float* o = out + (size_t)row * HIDD + h * HDIM + 2 * lane;
  o[0] = ax; o[1] = ay;
}

extern "C" void kernel_launch(void* const* d_in, const int* in_sizes, int n_in,
                              void* d_out, int out_size, void* d_ws, size_t ws_size,
                              hipStream_t stream) {
  (void)in_sizes; (void)n_in; (void)out_size; (void)ws_size;
  const float* hs  = (const float*)d_in[0];
  const float* K1w = (const float*)d_in[1];
  const float* K1b = (const float*)d_in[2];
  const float* V1w = (const float*)d_in[3];
  const float* V1b = (const float*)d_in[4];
  const float* RH  = (const float*)d_in[5];
  const float* bw  = (const float*)d_in[6];
  float* out = (float*)d_out;

  const size_t MROWS = (size_t)BSZ * LSEQ; // 8192
  float* K1 = (float*)d_ws;                // 8192*768 f32
  float* V1 = K1 + MROWS * HIDD;           // 8192*768 f32
  float* dp = V1 + MROWS * HIDD;           // 8192*12  f32
  int*  pos = (int*)(dp + MROWS * NH);     // 48*2048  i32
  int*  cnt = pos + BSZ * NH * LSEQ;       // 48       i32

  // 16 M-blocks (512 rows) x 24 N-blocks (32 cols) = 384 blocks, 8 waves each
  gemm_relu_wmma<<<384, 256, 0, stream>>>(hs, K1w, K1b, K1);
  gemm_relu_wmma<<<384, 256, 0, stream>>>(hs, V1w, V1b, V1);
  dp_kernel<<<1024, 256, 0, stream>>>(K1, RH, dp);         // 8192 rows / 8
  scan_kernel<<<1, 64, 0, stream>>>(dp, pos, cnt);         // 48 sequences
  gather_kernel<<<12288, 256, 0, stream>>>(V1, pos, cnt, bw, out); // 98304 waves / 8
}